// InertialCanonicalizer_67894843015434
// MI455X (gfx1250) — compile-verified
//
#include <hip/hip_runtime.h>
#include <math.h>

typedef float v2f __attribute__((ext_vector_type(2)));
typedef float v8f __attribute__((ext_vector_type(8)));

#define EPS_SIGN 1e-4f

#if defined(__has_builtin)
#if __has_builtin(__builtin_amdgcn_wmma_f32_16x16x4_f32)
#define HAVE_WMMA4 1
#endif
#endif

// ---------------- K0: per-graph start offsets via binary search on sorted batch ----
__global__ void k_starts(const int* __restrict__ batch, int n, int G,
                         int* __restrict__ start) {
  int g = blockIdx.x * blockDim.x + threadIdx.x;
  if (g > G) return;
  if (g == G) { start[G] = n; return; }
  int lo = 0, hi = n;
  while (lo < hi) {
    int mid = (lo + hi) >> 1;
    if (batch[mid] < g) lo = mid + 1; else hi = mid;
  }
  start[g] = lo;
}

// ---------------- K1: one wave32 per graph; homogeneous 4x4 Gram via V_WMMA_F32_16X16X4_F32
// D[m][n] (m,n<4) = sum_atoms [x y z 1]_m [x y z 1]_n  ->  S2, S1, count in one matrix.
__global__ __launch_bounds__(256) void k_moments(const float* __restrict__ pos,
                                                 const int* __restrict__ start,
                                                 int G, float* __restrict__ mom) {
  __shared__ float sm[8 * 96];
  const int wave = threadIdx.x >> 5;
  const int lane = threadIdx.x & 31;
  const int g = blockIdx.x * 8 + wave;
  if (g >= G) return;
  const int s = start[g];
  const int e = start[g + 1];
  float* smw = sm + wave * 96;
  const int m = lane & 15;        // A-matrix row (M) / B-matrix col (N) this lane holds
  const int half = lane >> 4;     // lanes 16-31 hold K=2,3
  const int mi = (m < 3) ? m : 0; // clamped LDS read index
  // lane-constant selection coefficients (branchless operand build)
  const float coefC = (m < 3) ? 1.0f : 0.0f;   // pass through coordinate
  const float isOne = (m == 3) ? 1.0f : 0.0f;  // homogeneous '1' row/col

#if HAVE_WMMA4
  v8f acc = {};
  for (int base = s; base < e; base += 32) {
    // stage a 32-atom tile (96 floats) into wave-private LDS, zero-padded past e.
    // clamped unconditional loads: no EXEC divergence in the hot loop.
    const int fb = base * 3, fe = e * 3;
#pragma unroll
    for (int j = 0; j < 3; ++j) {
      int idx = fb + lane + 32 * j;
      int idxc = (idx < fe) ? idx : (fe - 1);   // in-bounds: loop entered => fe >= 3
      float v = pos[idxc];
      smw[lane + 32 * j] = (idx < fe) ? v : 0.0f;
    }
    asm volatile("s_wait_dscnt 0x0" ::: "memory"); // per-wave LDS store->load ordering

#pragma unroll
    for (int w = 0; w < 8; ++w) {
      // f32 16x16x4 A layout: lanes0-15 VGPR0->K0, VGPR1->K1; lanes16-31 VGPR0->K2, VGPR1->K3.
      // B (4x16) mirrors it; our [p;1] construction is symmetric => B register image == A.
      const int a0 = 4 * w + 2 * half;     // atom for VGPR0
      const int a1 = a0 + 1;               // atom for VGPR1
      const float one0 = ((base + a0) < e) ? 1.0f : 0.0f;
      const float one1 = ((base + a1) < e) ? 1.0f : 0.0f;
      const float l0 = smw[a0 * 3 + mi];   // unconditional ds_load (zero-padded data)
      const float l1 = smw[a1 * 3 + mi];
      v2f A;
      A.x = fmaf(coefC, l0, isOne * one0); // rows/cols 4..15 become exactly 0
      A.y = fmaf(coefC, l1, isOne * one1);
      acc = __builtin_amdgcn_wmma_f32_16x16x4_f32(
          /*neg_a=*/false, A, /*neg_b=*/false, A,
          /*c_mod=*/(short)0, acc, /*reuse_a=*/false, /*reuse_b=*/false);
    }
  }
  // C/D layout: acc[r] on lane n (n<16) = D[r][n]. Lanes 0-3 write 4x4 column-major.
  if (lane < 4) {
    float4 col = make_float4(acc[0], acc[1], acc[2], acc[3]);
    *reinterpret_cast<float4*>(mom + (size_t)g * 16 + lane * 4) = col;
  }
#else
  // Fallback (no WMMA builtin): per-lane scalar moments + shuffle reduction
  float s10[10] = {0, 0, 0, 0, 0, 0, 0, 0, 0, 0};
  for (int base = s; base < e; base += 32) {
    int idx = base + lane;
    if (idx < e) {
      float x = pos[idx * 3], y = pos[idx * 3 + 1], z = pos[idx * 3 + 2];
      s10[0] += x * x; s10[1] += x * y; s10[2] += x * z;
      s10[3] += y * y; s10[4] += y * z; s10[5] += z * z;
      s10[6] += x; s10[7] += y; s10[8] += z; s10[9] += 1.0f;
    }
  }
#pragma unroll
  for (int k = 0; k < 10; ++k)
#pragma unroll
    for (int off = 16; off > 0; off >>= 1) s10[k] += __shfl_xor(s10[k], off, 32);
  if (lane == 0) {
    float* Mo = mom + (size_t)g * 16;
    Mo[0] = s10[0]; Mo[4] = s10[1]; Mo[8] = s10[2];
    Mo[1] = s10[1]; Mo[5] = s10[3]; Mo[9] = s10[4];
    Mo[2] = s10[2]; Mo[6] = s10[4]; Mo[10] = s10[5];
    Mo[3] = s10[6]; Mo[7] = s10[7]; Mo[11] = s10[8];
    Mo[12] = s10[6]; Mo[13] = s10[7]; Mo[14] = s10[8]; Mo[15] = s10[9];
  }
#endif
}

// ---------------- K2: one thread per graph; centroid, inertia, Jacobi eigensolve ----
__global__ __launch_bounds__(256) void k_eig(const float* __restrict__ mom, int G,
                                             float* __restrict__ center,
                                             float* __restrict__ Qout) {
  int g = blockIdx.x * blockDim.x + threadIdx.x;
  if (g >= G) return;
  const float* M = mom + (size_t)g * 16;  // column-major 4x4: M[col*4+row]
  float sxx = M[0], sxy = M[4], sxz = M[8];
  float syy = M[5], syz = M[9], szz = M[10];
  float s1x = M[12], s1y = M[13], s1z = M[14];
  float cnt = M[15];
  float inv = (cnt > 0.5f) ? (1.0f / cnt) : 0.0f;  // ref: divide by max(count,1)
  float cx = s1x * inv, cy = s1y * inv, cz = s1z * inv;
  // centered second moments:  sum pc pc^T = S2 - n c c^T
  float cxx = sxx - cnt * cx * cx, cyy = syy - cnt * cy * cy, czz = szz - cnt * cz * cz;
  float cxy = sxy - cnt * cx * cy, cxz = sxz - cnt * cx * cz, cyz = syz - cnt * cy * cz;
  float tr = cxx + cyy + czz;
  // inertia: tr*I - Cov
  float a00 = tr - cxx, a11 = tr - cyy, a22 = tr - czz;
  float a01 = -cxy, a02 = -cxz, a12 = -cyz;
  float v00 = 1.f, v01 = 0.f, v02 = 0.f;
  float v10 = 0.f, v11 = 1.f, v12 = 0.f;
  float v20 = 0.f, v21 = 0.f, v22 = 1.f;

#define JROT(app, aqq, apq, apr, aqr, vap, vaq, vbp, vbq, vcp, vcq)            \
  do {                                                                         \
    float _o = (apq);                                                          \
    if (fabsf(_o) > 1e-12f) {                                                  \
      float _th = ((aqq) - (app)) / (2.0f * _o);                               \
      float _t = 1.0f / (fabsf(_th) + sqrtf(_th * _th + 1.0f));                \
      _t = (_th < 0.0f) ? -_t : _t;                                            \
      float _c = rsqrtf(_t * _t + 1.0f);                                       \
      float _s = _t * _c;                                                      \
      (app) -= _t * _o; (aqq) += _t * _o; (apq) = 0.0f;                        \
      float _x = (apr), _y = (aqr);                                            \
      (apr) = _c * _x - _s * _y; (aqr) = _s * _x + _c * _y;                    \
      _x = (vap); (vap) = _c * _x - _s * (vaq); (vaq) = _s * _x + _c * (vaq);  \
      _x = (vbp); (vbp) = _c * _x - _s * (vbq); (vbq) = _s * _x + _c * (vbq);  \
      _x = (vcp); (vcp) = _c * _x - _s * (vcq); (vcq) = _s * _x + _c * (vcq);  \
    }                                                                          \
  } while (0)

#pragma unroll 1
  for (int sweep = 0; sweep < 10; ++sweep) {
    JROT(a00, a11, a01, a02, a12, v00, v01, v10, v11, v20, v21);  // (p,q)=(0,1)
    JROT(a00, a22, a02, a01, a12, v00, v02, v10, v12, v20, v22);  // (0,2)
    JROT(a11, a22, a12, a01, a02, v01, v02, v11, v12, v21, v22);  // (1,2)
  }
#undef JROT

  float l0 = a00, l1 = a11, l2 = a22;  // eigenvalues; sort ascending w/ column swaps
#define CSWAP(li, lj, x0, x1, y0, y1, z0, z1)                                   \
  if ((lj) < (li)) {                                                            \
    float _t;                                                                   \
    _t = (li); (li) = (lj); (lj) = _t;                                          \
    _t = (x0); (x0) = (x1); (x1) = _t;                                          \
    _t = (y0); (y0) = (y1); (y1) = _t;                                          \
    _t = (z0); (z0) = (z1); (z1) = _t;                                          \
  }
  CSWAP(l0, l1, v00, v01, v10, v11, v20, v21);
  CSWAP(l1, l2, v01, v02, v11, v12, v21, v22);
  CSWAP(l0, l1, v00, v01, v10, v11, v20, v21);
#undef CSWAP

  center[(size_t)g * 3 + 0] = cx;
  center[(size_t)g * 3 + 1] = cy;
  center[(size_t)g * 3 + 2] = cz;
  float* Q = Qout + (size_t)g * 9;  // row-major Q[i][j], columns = eigenvectors
  Q[0] = v00; Q[1] = v01; Q[2] = v02;
  Q[3] = v10; Q[4] = v11; Q[5] = v12;
  Q[6] = v20; Q[7] = v21; Q[8] = v22;
}

// ---------------- K3: one wave per graph; early-exit sign scan + output assembly ----
__global__ __launch_bounds__(256) void k_sign(const float* __restrict__ pos,
                                              const int* __restrict__ start,
                                              const float* __restrict__ center,
                                              const float* __restrict__ Qp,
                                              int G, float* __restrict__ out) {
  const int wave = threadIdx.x >> 5;
  const int lane = threadIdx.x & 31;
  const int g = blockIdx.x * 8 + wave;
  if (g >= G) return;
  const int s = start[g], e = start[g + 1];
  const float cx = center[g * 3 + 0], cy = center[g * 3 + 1], cz = center[g * 3 + 2];
  const float* Q = Qp + (size_t)g * 9;
  const float q00 = Q[0], q01 = Q[1];
  const float q10 = Q[3], q11 = Q[4];
  const float q20 = Q[6], q21 = Q[7];
  float s0 = 0.0f, s1 = 0.0f;  // 0 = not found yet

  for (int base = s; base < e && (s0 == 0.0f || s1 == 0.0f); base += 32) {
    int idx = base + lane;
    bool valid = idx < e;
    int idxc = valid ? idx : (e - 1);  // clamped, unconditional coalesced load
    float px = pos[idxc * 3 + 0] - cx;
    float py = pos[idxc * 3 + 1] - cy;
    float pz = pos[idxc * 3 + 2] - cz;
    float p0 = px * q00 + py * q10 + pz * q20;
    float p1 = px * q01 + py * q11 + pz * q21;
#if defined(__has_builtin) && __has_builtin(__builtin_amdgcn_ballot_w32)
    unsigned m0 = __builtin_amdgcn_ballot_w32(valid && (fabsf(p0) > EPS_SIGN));
    unsigned m1 = __builtin_amdgcn_ballot_w32(valid && (fabsf(p1) > EPS_SIGN));
#else
    unsigned m0 = (unsigned)__ballot(valid && (fabsf(p0) > EPS_SIGN));
    unsigned m1 = (unsigned)__ballot(valid && (fabsf(p1) > EPS_SIGN));
#endif
    if (s0 == 0.0f && m0) {
      int l = __builtin_ctz(m0);               // lowest index qualifying atom
      float v = __shfl(p0, l, 32);
      s0 = (v < 0.0f) ? -1.0f : 1.0f;
    }
    if (s1 == 0.0f && m1) {
      int l = __builtin_ctz(m1);
      float v = __shfl(p1, l, 32);
      s1 = (v < 0.0f) ? -1.0f : 1.0f;
    }
  }
  if (s0 == 0.0f) s0 = 1.0f;
  if (s1 == 0.0f) s1 = 1.0f;

  if (lane == 0) {
    float q0x = s0 * q00, q0y = s0 * q10, q0z = s0 * q20;
    float q1x = s1 * q01, q1y = s1 * q11, q1z = s1 * q21;
    float q2x = q0y * q1z - q0z * q1y;
    float q2y = q0z * q1x - q0x * q1z;
    float q2z = q0x * q1y - q0y * q1x;
    float* o = out + (size_t)g * 9;  // out[g,i,j]: columns j = (q0,q1,q2)
    o[0] = q0x; o[1] = q1x; o[2] = q2x;
    o[3] = q0y; o[4] = q1y; o[5] = q2y;
    o[6] = q0z; o[7] = q1z; o[8] = q2z;
  }
}

extern "C" void kernel_launch(void* const* d_in, const int* in_sizes, int n_in,
                              void* d_out, int out_size, void* d_ws, size_t ws_size,
                              hipStream_t stream) {
  (void)n_in; (void)ws_size;
  const float* pos = (const float*)d_in[0];
  const int* batch = (const int*)d_in[1];
  const int N = in_sizes[0] / 3;
  const int G = out_size / 9;

  // scratch layout (~3.8 MB): start[G+1] | mom[G*16] | center[G*3] | Qpre[G*9]
  char* ws = (char*)d_ws;
  int* start = (int*)ws;
  size_t off = (((size_t)(G + 1) * sizeof(int)) + 15) & ~(size_t)15;
  float* mom = (float*)(ws + off); off += (size_t)G * 16 * sizeof(float);
  float* center = (float*)(ws + off); off += (size_t)G * 3 * sizeof(float);
  float* Qpre = (float*)(ws + off);
  float* out = (float*)d_out;

  k_starts<<<(G + 1 + 255) / 256, 256, 0, stream>>>(batch, N, G, start);
  k_moments<<<(G + 7) / 8, 256, 0, stream>>>(pos, start, G, mom);
  k_eig<<<(G + 255) / 256, 256, 0, stream>>>(mom, G, center, Qpre);
  k_sign<<<(G + 7) / 8, 256, 0, stream>>>(pos, start, center, Qpre, G, out);
}